// GraphDecoder_39479339384908
// MI455X (gfx1250) — compile-verified
//
#include <hip/hip_runtime.h>
#include <hip/hip_bf16.h>

typedef __attribute__((ext_vector_type(2))) float v2f;
typedef __attribute__((ext_vector_type(8))) float v8f;

#define N_NODES 20000
#define BATCH   8

// ---------------------------------------------------------------------------
// Stage 1: h = z @ w_proj + b_proj   (M=8 padded to 16, K=64, N=640000)
// One wave per 16-column tile; V_WMMA_F32_16X16X4_F32, 16 k-steps.
// z is staged zero-padded in LDS (16x64) so the inner loop has no predication.
// ---------------------------------------------------------------------------
__global__ __launch_bounds__(256) void proj_kernel(
    const float* __restrict__ z,      // [8,64]
    const float* __restrict__ wp,     // [64,640000]
    const float* __restrict__ bp,     // [640000]
    float* __restrict__ out,          // [8,640000]  == [B,N,32] flat
    int ncols)
{
  __shared__ float zpad[16 * 64];                 // rows 8..15 zero
  for (int i = threadIdx.x; i < 16 * 64; i += 256) {
    const int row = i >> 6, k = i & 63;
    zpad[i] = (row < BATCH) ? z[row * 64 + k] : 0.0f;
  }
  __syncthreads();

  const int lane = threadIdx.x & 31;
  const int wave = threadIdx.x >> 5;
  const int tile = blockIdx.x * 8 + wave;         // 40000 tiles total
  const long col0 = (long)tile * 16;
  const int m   = lane & 15;                      // A row / B col / C col
  const int klo = (lane >> 4) * 2;                // k sub-pair per half-wave

  v8f c = {};
  #pragma unroll
  for (int kb = 0; kb < 64; kb += 4) {
    v2f a = *(const v2f*)&zpad[m * 64 + kb + klo];     // ds_load_b64, no predication
    v2f b;
    b.x = wp[(long)(kb + klo)     * ncols + col0 + m];
    b.y = wp[(long)(kb + klo + 1) * ncols + col0 + m];
    c = __builtin_amdgcn_wmma_f32_16x16x4_f32(false, a, false, b,
                                              (short)0, c, false, false);
  }
  const float bias = bp[col0 + m];
  #pragma unroll
  for (int j = 0; j < 8; ++j) {
    const int row = j + ((lane >> 4) << 3);       // C layout: lanes>=16 hold rows 8..15
    if (row < BATCH)
      out[(long)row * ncols + col0 + m] = c[j] + bias;
  }
}

// ---------------------------------------------------------------------------
// Elementwise init: dst = alpha * src   (alpha=0 -> zero, alpha=-1 -> -T_{k-2})
// ---------------------------------------------------------------------------
__global__ void axpy_init(float* __restrict__ dst, const float* __restrict__ src,
                          float alpha, long n)
{
  long i = (long)blockIdx.x * blockDim.x + threadIdx.x;
  if (i < n) dst[i] = alpha * src[i];
}

// ---------------------------------------------------------------------------
// SpMM over COO edges (float4-vectorized along features):
//   dst[b, rows[e], f..f+3] += scale * lvals[e] * x[b, cols[e], f..f+3]
// ---------------------------------------------------------------------------
template<int F>
__global__ void spmm_kernel(float* __restrict__ dst, const float* __restrict__ x,
                            const int* __restrict__ rows, const int* __restrict__ cols,
                            const float* __restrict__ lvals, float scale, int E)
{
  constexpr int G = F / 4;                        // float4 groups per node row
  long tid = (long)blockIdx.x * blockDim.x + threadIdx.x;
  if (tid >= (long)E * G) return;
  const int e = (int)(tid / G);
  const int f = 4 * (int)(tid - (long)e * G);
  const int r = rows[e];
  const int cidx = cols[e];
  const float v = scale * lvals[e];
  const long plane = (long)N_NODES * F;
  #pragma unroll
  for (int b = 0; b < BATCH; ++b) {
    const float4 xv = *(const float4*)(x + (long)b * plane + (long)cidx * F + f);
    float* d = dst + (long)b * plane + (long)r * F + f;
    __hip_atomic_fetch_add(d + 0, xv.x * v, __ATOMIC_RELAXED, __HIP_MEMORY_SCOPE_AGENT);
    __hip_atomic_fetch_add(d + 1, xv.y * v, __ATOMIC_RELAXED, __HIP_MEMORY_SCOPE_AGENT);
    __hip_atomic_fetch_add(d + 2, xv.z * v, __ATOMIC_RELAXED, __HIP_MEMORY_SCOPE_AGENT);
    __hip_atomic_fetch_add(d + 3, xv.w * v, __ATOMIC_RELAXED, __HIP_MEMORY_SCOPE_AGENT);
  }
}

// ---------------------------------------------------------------------------
// Combine: dst[r, fo] = elu?( sum_{k,fi} T[k, r, fi] * w[k, fi, fo] + bias[fo] )
// R = B*N rows, one 16-row tile per wave; f32 WMMA 16x16x4 over KK = 5*Fin.
// w is staged in LDS zero-padded to 16 columns and repacked as (kk,kk+1) v2f
// pairs, so B fetches are single unpredicated ds_load_b64s. A fetches are
// contiguous-pair global_load_b64s.
// ---------------------------------------------------------------------------
template<int Fin, int Fout, bool DO_ELU>
__global__ __launch_bounds__(256) void cheb_combine(
    const float* __restrict__ T,      // [5, R, Fin]
    const float* __restrict__ w,      // [5, Fin, Fout]
    const float* __restrict__ bias,   // [Fout]
    float* __restrict__ dst,          // [R, Fout]
    int R)
{
  constexpr int KK = 5 * Fin;
  __shared__ v2f wpad[(KK / 2) * 16];             // [pair][col], cols>=Fout are 0
  for (int i = threadIdx.x; i < (KK / 2) * 16; i += 256) {
    const int pair = i >> 4, col = i & 15;
    const int kk0 = 2 * pair, kk1 = kk0 + 1;
    v2f val;
    val.x = (col < Fout) ? w[(kk0 / Fin) * Fin * Fout + (kk0 % Fin) * Fout + col] : 0.0f;
    val.y = (col < Fout) ? w[(kk1 / Fin) * Fin * Fout + (kk1 % Fin) * Fout + col] : 0.0f;
    wpad[i] = val;
  }
  __syncthreads();

  const int lane = threadIdx.x & 31;
  const int wave = threadIdx.x >> 5;
  const int tile = blockIdx.x * 8 + wave;         // R/16 tiles
  const int m   = lane & 15;
  const int klo = (lane >> 4) * 2;
  const long r = (long)tile * 16 + m;             // A row index
  const long plane = (long)R * Fin;

  v8f c = {};
  #pragma unroll
  for (int kb = 0; kb < KK; kb += 4) {
    const int kk0 = kb + klo;                     // even; kk0,kk0+1 share a fin row
    const v2f a = *(const v2f*)(T + (long)(kk0 / Fin) * plane
                                  + r * Fin + (kk0 % Fin));   // global_load_b64
    const v2f b = wpad[(kk0 >> 1) * 16 + m];                  // ds_load_b64
    c = __builtin_amdgcn_wmma_f32_16x16x4_f32(false, a, false, b,
                                              (short)0, c, false, false);
  }
  const float bv = (m < Fout) ? bias[m] : 0.0f;
  #pragma unroll
  for (int j = 0; j < 8; ++j) {
    const long row = (long)tile * 16 + j + ((lane >> 4) << 3);
    if (m < Fout) {
      float v = c[j] + bv;
      if (DO_ELU) v = (v > 0.0f) ? v : (__expf(v) - 1.0f);
      dst[row * Fout + m] = v;
    }
  }
}

// ---------------------------------------------------------------------------
// Host-side: Chebyshev stack builder  T1 = L x ; Tk = 2 L T_{k-1} - T_{k-2}
// ---------------------------------------------------------------------------
template<int F>
static void run_cheb_stack(float* S, const int* rows, const int* cols,
                           const float* lvals, int E, hipStream_t stream)
{
  const long szk = (long)BATCH * N_NODES * F;     // floats per T_k
  const int ablocks = (int)((szk + 255) / 256);
  const int eblocks = (int)(((long)E * (F / 4) + 255) / 256);
  // T1 = L @ T0
  axpy_init<<<ablocks, 256, 0, stream>>>(S + szk, S, 0.0f, szk);
  spmm_kernel<F><<<eblocks, 256, 0, stream>>>(S + szk, S, rows, cols, lvals, 1.0f, E);
  // Tk = 2 L T_{k-1} - T_{k-2}
  for (int k = 2; k <= 4; ++k) {
    axpy_init<<<ablocks, 256, 0, stream>>>(S + (long)k * szk, S + (long)(k - 2) * szk,
                                           -1.0f, szk);
    spmm_kernel<F><<<eblocks, 256, 0, stream>>>(S + (long)k * szk, S + (long)(k - 1) * szk,
                                                rows, cols, lvals, 2.0f, E);
  }
}

extern "C" void kernel_launch(void* const* d_in, const int* in_sizes, int n_in,
                              void* d_out, int out_size, void* d_ws, size_t ws_size,
                              hipStream_t stream)
{
  const float* z      = (const float*)d_in[0];   // [8,64]
  const float* w_proj = (const float*)d_in[1];   // [64,640000]
  const float* b_proj = (const float*)d_in[2];   // [640000]
  const float* w1     = (const float*)d_in[3];   // [5,32,16]
  const float* b1     = (const float*)d_in[4];   // [16]
  const float* w2     = (const float*)d_in[5];   // [5,16,8]
  const float* b2     = (const float*)d_in[6];   // [8]
  const float* w3     = (const float*)d_in[7];   // [5,8,3]
  const float* b3     = (const float*)d_in[8];   // [3]
  const int*   rows   = (const int*)d_in[9];     // [E]
  const int*   cols   = (const int*)d_in[10];    // [E]
  const float* lvals  = (const float*)d_in[11];  // [E]
  const int E = in_sizes[9];                     // 320000

  const int R = BATCH * N_NODES;                 // 160000 GEMM rows
  float* stackA = (float*)d_ws;                            // 5*[B,N,32] = 102.4 MB
  const long szA = (long)BATCH * N_NODES * 32;
  float* stackB = stackA + 5 * szA;                        // 5*[B,N,16] =  51.2 MB
  float* stackC = stackA;                                  // reuse dead region A

  // Stage 1: projection (bandwidth-bound stream of w_proj, f32 WMMA)
  proj_kernel<<<5000, 256, 0, stream>>>(z, w_proj, b_proj, stackA, N_NODES * 32);

  // Layer 1: Fin=32 -> Fout=16, ELU
  run_cheb_stack<32>(stackA, rows, cols, lvals, E, stream);
  cheb_combine<32, 16, true><<<R / 16 / 8, 256, 0, stream>>>(stackA, w1, b1, stackB, R);

  // Layer 2: Fin=16 -> Fout=8, ELU
  run_cheb_stack<16>(stackB, rows, cols, lvals, E, stream);
  cheb_combine<16, 8, true><<<R / 16 / 8, 256, 0, stream>>>(stackB, w2, b2, stackC, R);

  // Layer 3: Fin=8 -> Fout=3, no activation, straight to d_out
  run_cheb_stack<8>(stackC, rows, cols, lvals, E, stream);
  cheb_combine<8, 3, false><<<R / 16 / 8, 256, 0, stream>>>(stackC, w3, b3,
                                                            (float*)d_out, R);
}